// HungarianMatcher_dotproduct_18313740550408
// MI455X (gfx1250) — compile-verified
//
#include <hip/hip_runtime.h>
#include <math.h>

typedef __attribute__((ext_vector_type(2))) float v2f;
typedef __attribute__((ext_vector_type(8))) float v8f;

#define B_    8
#define Q_    512
#define G_    32
#define MCOLS 512        // queries (columns in JV after transpose)
#define NROWS 32         // golds   (rows in JV)
#define NEGBIG (-1000000000.0f)
#define LOG2E  1.4426950408889634f

__device__ __forceinline__ float fix_nan(float x) { return (x != x) ? NEGBIG : x; }

// ---------------------------------------------------------------------------
// Kernel 1: per-row (b,q) compute m, S2 for all 5 tensors, gather 32 gold
// columns, emit cost[b,q,g].  One wave per row; lane g owns gold slot g.
// ---------------------------------------------------------------------------
__global__ __launch_bounds__(256) void cost_kernel(
    const float* __restrict__ rel, const float* __restrict__ hs,
    const float* __restrict__ he,  const float* __restrict__ ts,
    const float* __restrict__ te,
    const int* __restrict__ grel,  const int* __restrict__ ghs,
    const int* __restrict__ ghe,   const int* __restrict__ gts,
    const int* __restrict__ gte,
    float* __restrict__ cost)
{
    const int lane = threadIdx.x & 31;
    const int wave = threadIdx.x >> 5;
    const int row  = blockIdx.x * 8 + wave;   // b*Q + q, 0..4095
    const int b    = row >> 9;

    const float* bases[5] = { rel, hs, he, ts, te };
    const int*   golds[5] = { grel, ghs, ghe, gts, gte };
    const int    lens[5]  = { 512, 1024, 1024, 1024, 1024 };

    float acc = 0.0f;
#pragma unroll
    for (int t = 0; t < 5; ++t) {
        const int L = lens[t];
        const float* rowp = bases[t] + (size_t)row * (size_t)L;
        if (t < 4)  // warm next tensor's row (global_prefetch_b8)
            __builtin_prefetch(bases[t + 1] + (size_t)row * (size_t)lens[t + 1], 0, 1);

        const float4* p4 = (const float4*)rowp;
        const int n4 = L >> 7;                 // float4s per lane: 4 or 8

        // ---- pass A: row max (HBM pass) ----
        float m = -INFINITY;
        for (int i = 0; i < n4; ++i) {
            float4 x = p4[i * 32 + lane];
            m = fmaxf(m, fmaxf(fmaxf(fix_nan(x.x), fix_nan(x.y)),
                               fmaxf(fix_nan(x.z), fix_nan(x.w))));
        }
#pragma unroll
        for (int off = 16; off; off >>= 1)
            m = fmaxf(m, __shfl_xor(m, off, 32));

        // ---- pass B: per-lane partial of S2 = sum exp(2*(x-m)) (L2-hot) ----
        float s = 0.0f;
        for (int i = 0; i < n4; ++i) {
            float4 x = p4[i * 32 + lane];
            s += exp2f((fix_nan(x.x) - m) * (2.0f * LOG2E));
            s += exp2f((fix_nan(x.y) - m) * (2.0f * LOG2E));
            s += exp2f((fix_nan(x.z) - m) * (2.0f * LOG2E));
            s += exp2f((fix_nan(x.w) - m) * (2.0f * LOG2E));
        }

        // ---- 32-lane sum via WMMA: D = A(16x4) * ones(4x16) + 0 ----
        // A layout: lane<16 -> A[lane][0]=s, lane>=16 -> A[lane-16][2]=s,
        // so D[m][n] = s_m + s_{m+16} for every n.
        v2f a;   a.x = s;    a.y = 0.0f;
        v2f one; one.x = 1.0f; one.y = 1.0f;
        v8f d = {};
        d = __builtin_amdgcn_wmma_f32_16x16x4_f32(false, a, false, one,
                                                  (short)0, d, false, false);
        float half = d[0] + d[1] + d[2] + d[3] + d[4] + d[5] + d[6] + d[7];
        float s2 = half + __shfl_xor(half, 16, 32);   // exact 32-lane total

        // ---- gather: lane g handles gold slot g ----
        int   gi = golds[t][b * G_ + lane];
        float xg = fix_nan(rowp[gi]);
        float w  = (t == 0) ? 1.0f : 0.5f;            // W_REL=1, 0.5*W_HEAD/TAIL
        acc += w * exp2f((xg - m) * LOG2E) * rsqrtf(s2);
    }

    cost[(size_t)row * G_ + lane] = -acc;             // coalesced 128B store
}

// ---------------------------------------------------------------------------
// Kernel 2: exact JV shortest-augmenting-path assignment, one block per batch.
// C^T[i][j] = cost[b][j][i]  (i = gold 1..32, j = query 1..512)
// ---------------------------------------------------------------------------
__global__ __launch_bounds__(512) void hungarian_kernel(
    const float* __restrict__ cost, float* __restrict__ rowOut,
    float* __restrict__ colOut)
{
    const int b   = blockIdx.x;
    const int tid = threadIdx.x;
    const float INFv = __builtin_inff();
    const float* Cb = cost + (size_t)b * MCOLS * NROWS;  // [j][i]

    __shared__ float u_[NROWS + 1];
    __shared__ float v_[MCOLS + 1];
    __shared__ float minv[MCOLS + 1];
    __shared__ int   way[MCOLS + 1];
    __shared__ int   p_[MCOLS + 1];
    __shared__ unsigned int used[MCOLS + 1];
    __shared__ float wmin[16];
    __shared__ int   widx[16];
    __shared__ int   j0_s;
    __shared__ float delta_sh;
    __shared__ int   rows_s[NROWS];

    for (int j = tid; j <= MCOLS; j += 512) { v_[j] = 0.0f; p_[j] = 0; way[j] = 0; }
    if (tid <= NROWS) u_[tid] = 0.0f;
    __syncthreads();

    for (int i = 1; i <= NROWS; ++i) {
        if (tid == 0) { p_[0] = i; j0_s = 0; }
        for (int j = tid; j <= MCOLS; j += 512) { minv[j] = INFv; used[j] = 0; }
        __syncthreads();

        while (true) {
            if (tid == 0) used[j0_s] = 1;
            __syncthreads();                          // (A)
            const int j0 = j0_s;
            const int i0 = p_[j0];

            // scan my column j = tid+1 (1..512)
            const int j = tid + 1;
            float mv; int bi = j;
            if (!used[j]) {
                float cur = Cb[(size_t)(j - 1) * NROWS + (i0 - 1)] - u_[i0] - v_[j];
                if (cur < minv[j]) { minv[j] = cur; way[j] = j0; }
                mv = minv[j];
            } else {
                mv = INFv;
            }

            // argmin stage 1: within each wave32, ties -> lowest j
#pragma unroll
            for (int off = 16; off; off >>= 1) {
                float ov = __shfl_xor(mv, off, 32);
                int   oi = __shfl_xor(bi, off, 32);
                if (ov < mv || (ov == mv && oi < bi)) { mv = ov; bi = oi; }
            }
            if ((tid & 31) == 0) { wmin[tid >> 5] = mv; widx[tid >> 5] = bi; }
            __syncthreads();                          // (B)

            // argmin stage 2: 16 wave results reduced in wave 0
            if (tid < 16) {
                mv = wmin[tid]; bi = widx[tid];
#pragma unroll
                for (int off = 8; off; off >>= 1) {
                    float ov = __shfl_xor(mv, off, 32);
                    int   oi = __shfl_xor(bi, off, 32);
                    if (ov < mv || (ov == mv && oi < bi)) { mv = ov; bi = oi; }
                }
                if (tid == 0) { delta_sh = mv; j0_s = bi; }
            }
            __syncthreads();                          // (C)

            const float delta = delta_sh;
            const int   j1    = j0_s;
            for (int jj = tid; jj <= MCOLS; jj += 512) {
                if (used[jj]) { u_[p_[jj]] += delta; v_[jj] -= delta; }
                else          { minv[jj] -= delta; }
            }
            __syncthreads();                          // (D)
            if (p_[j1] == 0) break;
        }

        if (tid == 0) {                               // augment along path
            int j0 = j0_s;
            while (j0) { int jp = way[j0]; p_[j0] = p_[jp]; j0 = jp; }
        }
        __syncthreads();
    }

    for (int j = tid + 1; j <= MCOLS; j += 512) {
        int pi = p_[j];
        if (pi > 0) rows_s[pi - 1] = j - 1;
    }
    __syncthreads();

    if (tid == 0) {                                   // scipy convention sort
        int rr[NROWS], cc[NROWS];
        for (int k = 0; k < NROWS; ++k) { rr[k] = rows_s[k]; cc[k] = k; }
        for (int a = 1; a < NROWS; ++a) {
            int kr = rr[a], kc = cc[a], e = a - 1;
            while (e >= 0 && rr[e] > kr) { rr[e + 1] = rr[e]; cc[e + 1] = cc[e]; --e; }
            rr[e + 1] = kr; cc[e + 1] = kc;
        }
        for (int k = 0; k < NROWS; ++k) {
            rowOut[b * NROWS + k] = (float)rr[k];
            colOut[b * NROWS + k] = (float)cc[k];
        }
    }
}

// ---------------------------------------------------------------------------
extern "C" void kernel_launch(void* const* d_in, const int* in_sizes, int n_in,
                              void* d_out, int out_size, void* d_ws, size_t ws_size,
                              hipStream_t stream)
{
    (void)in_sizes; (void)n_in; (void)out_size; (void)d_ws; (void)ws_size;

    const float* rel  = (const float*)d_in[0];
    const float* hs   = (const float*)d_in[1];
    const float* he   = (const float*)d_in[2];
    const float* ts   = (const float*)d_in[3];
    const float* te   = (const float*)d_in[4];
    const int*   grel = (const int*)d_in[5];   // jax x64 off: int64 -> int32
    const int*   ghs  = (const int*)d_in[6];
    const int*   ghe  = (const int*)d_in[7];
    const int*   gts  = (const int*)d_in[8];
    const int*   gte  = (const int*)d_in[9];

    float* out   = (float*)d_out;
    float* cost  = out;                                 // 8*512*32 floats
    float* rowi  = out + (size_t)B_ * Q_ * G_;          // 8*32
    float* coli  = rowi + B_ * G_;                      // 8*32

    // 4096 rows, 8 waves (256 threads) per block -> 512 blocks
    cost_kernel<<<dim3((B_ * Q_) / 8), dim3(256), 0, stream>>>(
        rel, hs, he, ts, te, grel, ghs, ghe, gts, gte, cost);

    hungarian_kernel<<<dim3(B_), dim3(512), 0, stream>>>(cost, rowi, coli);
}